// EnhancedGATBlock_41996190221097
// MI455X (gfx1250) — compile-verified
//
#include <hip/hip_runtime.h>
#include <math.h>

#define N_NODES 50000
#define N_EDGES 800000
#define IN_DIM 64
#define HID 64
#define HEADS 4
#define EDGE_DIM 32
#define FDIM (HEADS * HID)   // 256
#define NEG_SLOPE 0.2f
#define LN_EPS 1e-5f

typedef __attribute__((ext_vector_type(2))) float v2f;
typedef __attribute__((ext_vector_type(8))) float v8f;

// -------- float atomic max via monotonic int encoding --------
__device__ __forceinline__ void atomicMaxFloat(float* addr, float val) {
    if (val >= 0.0f) {
        atomicMax((int*)addr, __float_as_int(val));
    } else {
        atomicMin((unsigned int*)addr, (unsigned int)__float_as_int(val));
    }
}

// -------- 0: init accumulators --------
__global__ void gat_init_kernel(float* __restrict__ out_acc,
                                float* __restrict__ amax,
                                float* __restrict__ denom) {
    int idx = blockIdx.x * blockDim.x + threadIdx.x;
    if (idx < N_NODES * HID) out_acc[idx] = 0.0f;
    if (idx < N_NODES * HEADS) {
        amax[idx]  = -INFINITY;   // bits 0xFF800000: valid floor for both atomic paths
        denom[idx] = 0.0f;
    }
}

// -------- 1: node projections xl = x@W_l, xr = x@W_r (WMMA f32 16x16x4) --------
// one wave per 16x16 output tile; grid.y in [0,32): y<16 -> W_l, else W_r
__global__ void gat_proj_kernel(const float* __restrict__ x,
                                const float* __restrict__ Wl,
                                const float* __restrict__ Wr,
                                float* __restrict__ xl,
                                float* __restrict__ xr) {
    const int lane = threadIdx.x;      // 0..31 (wave32)
    const int m0   = blockIdx.x * 16;  // node row tile
    int y = blockIdx.y;
    const float* __restrict__ W;
    float* __restrict__ out;
    if (y < 16) { W = Wl; out = xl; } else { W = Wr; out = xr; y -= 16; }
    const int n0   = y * 16;           // column tile
    const int half = lane >> 4;        // 0: lanes 0-15, 1: lanes 16-31
    const int l15  = lane & 15;
    const int row  = m0 + l15;         // A-fragment row (per-lane)
    const int col  = n0 + l15;         // B/C/D column (per-lane)

    v8f acc = {};
    #pragma unroll
    for (int kk = 0; kk < IN_DIM; kk += 4) {
        const int ka = kk + half * 2;  // ISA A-layout: lanes 16-31 hold K+2,K+3
        v2f a, b;
        a.x = x[row * IN_DIM + ka];
        a.y = x[row * IN_DIM + ka + 1];
        b.x = W[(ka    ) * FDIM + col];
        b.y = W[(ka + 1) * FDIM + col];
        acc = __builtin_amdgcn_wmma_f32_16x16x4_f32(false, a, false, b,
                                                    (short)0, acc, false, false);
    }
    // D layout: VGPR i = row (m0 + half*8 + i), column = col
    #pragma unroll
    for (int i = 0; i < 8; ++i) {
        out[(m0 + half * 8 + i) * FDIM + col] = acc[i];
    }
}

// -------- 2: fused edge GEMM (edge_attr@W_e) + leaky-relu + att dot -> logits, atomic max --------
// one wave per (16-edge tile, head): 4 WMMA C-tiles = 64 columns of one head
__global__ void gat_edge_logits_kernel(const float* __restrict__ ea,   // [E,32]
                                       const float* __restrict__ We,   // [32,256]
                                       const float* __restrict__ xl,   // [N,256]
                                       const float* __restrict__ xr,   // [N,256]
                                       const float* __restrict__ att,  // [4,64]
                                       const int*   __restrict__ src,
                                       const int*   __restrict__ dst,
                                       float* __restrict__ logits,     // [E,4]
                                       float* __restrict__ amax) {     // [N,4]
    const int lane = threadIdx.x;
    const int e0   = blockIdx.x * 16;
    const int h    = blockIdx.y;
    const int half = lane >> 4;
    const int l15  = lane & 15;

    v8f acc[4] = {v8f{}, v8f{}, v8f{}, v8f{}};
    #pragma unroll
    for (int kk = 0; kk < EDGE_DIM; kk += 4) {
        const int ka = kk + half * 2;
        v2f a;
        a.x = ea[(e0 + l15) * EDGE_DIM + ka];
        a.y = ea[(e0 + l15) * EDGE_DIM + ka + 1];
        #pragma unroll
        for (int t = 0; t < 4; ++t) {
            const int col = h * HID + t * 16 + l15;
            v2f b;
            b.x = We[(ka    ) * FDIM + col];
            b.y = We[(ka + 1) * FDIM + col];
            acc[t] = __builtin_amdgcn_wmma_f32_16x16x4_f32(false, a, false, b,
                                                           (short)0, acc[t], false, false);
        }
    }

    // VGPR i of a C-tile = edge (e0 + half*8 + i); lane's column = t*16 + l15 within head
    #pragma unroll
    for (int i = 0; i < 8; ++i) {
        const int e = e0 + half * 8 + i;
        const int s = src[e];
        const int d = dst[e];
        float red = 0.0f;
        #pragma unroll
        for (int t = 0; t < 4; ++t) {
            const int cwh = t * 16 + l15;      // column within head
            const int col = h * HID + cwh;
            float z = acc[t][i] + xl[s * FDIM + col] + xr[d * FDIM + col];
            z = (z > 0.0f) ? z : NEG_SLOPE * z;
            red += z * att[h * HID + cwh];
        }
        // sum across the 16 lanes of each half-wave
        red += __shfl_xor(red, 1, 32);
        red += __shfl_xor(red, 2, 32);
        red += __shfl_xor(red, 4, 32);
        red += __shfl_xor(red, 8, 32);
        if (l15 == 0) {
            logits[e * HEADS + h] = red;
            atomicMaxFloat(&amax[d * HEADS + h], red);
        }
    }
}

// -------- 3: ex = exp(logit - amax[dst]); denom += ex (in-place over logits buffer) --------
__global__ void gat_exp_kernel(const int* __restrict__ dst,
                               const float* __restrict__ amax,
                               float* __restrict__ logits_ex,  // [E,4] in-place
                               float* __restrict__ denom) {    // [N,4]
    int idx = blockIdx.x * blockDim.x + threadIdx.x;
    if (idx >= N_EDGES * HEADS) return;
    const int e = idx >> 2;
    const int h = idx & 3;
    const int d = dst[e];
    float v = expf(logits_ex[idx] - amax[d * HEADS + h]);
    logits_ex[idx] = v;
    atomicAdd(&denom[d * HEADS + h], v);
}

// -------- 4: scatter-aggregate with folded head-mean: out[dst,c] += sum_h alpha*xl/4 --------
// one wave per edge (2 channels per lane)
__global__ void gat_aggregate_kernel(const float* __restrict__ xl,
                                     const float* __restrict__ ex,     // [E,4]
                                     const float* __restrict__ denom,  // [N,4]
                                     const int*   __restrict__ src,
                                     const int*   __restrict__ dst,
                                     float* __restrict__ out_acc) {    // [N,64]
    const int lane = threadIdx.x & 31;
    const int wid  = threadIdx.x >> 5;
    const int e    = blockIdx.x * 8 + wid;
    if (e >= N_EDGES) return;
    const int s = src[e];
    const int d = dst[e];
    float alpha[HEADS];
    #pragma unroll
    for (int h = 0; h < HEADS; ++h)
        alpha[h] = 0.25f * ex[e * HEADS + h] / denom[d * HEADS + h];
    #pragma unroll
    for (int cc = 0; cc < 2; ++cc) {
        const int c = lane + cc * 32;
        float v = 0.0f;
        #pragma unroll
        for (int h = 0; h < HEADS; ++h)
            v += alpha[h] * xl[s * FDIM + h * HID + c];
        atomicAdd(&out_acc[d * HID + c], v);
    }
}

// -------- 5: bias + residual + LayerNorm (in-place, one wave per node) --------
__global__ void gat_finalize_kernel(const float* __restrict__ x,
                                    const float* __restrict__ bias,
                                    const float* __restrict__ ln_w,
                                    const float* __restrict__ ln_b,
                                    float* __restrict__ out) {   // [N,64] in-place
    const int lane = threadIdx.x & 31;
    const int wid  = threadIdx.x >> 5;
    const int n    = blockIdx.x * 8 + wid;
    if (n >= N_NODES) return;
    const int c0 = lane, c1 = lane + 32;
    float h0 = out[n * HID + c0] + bias[c0] + x[n * HID + c0];
    float h1 = out[n * HID + c1] + bias[c1] + x[n * HID + c1];
    float sum = h0 + h1;
    #pragma unroll
    for (int m = 16; m >= 1; m >>= 1) sum += __shfl_xor(sum, m, 32);
    const float mu = sum * (1.0f / 64.0f);
    const float d0 = h0 - mu, d1 = h1 - mu;
    float vs = d0 * d0 + d1 * d1;
    #pragma unroll
    for (int m = 16; m >= 1; m >>= 1) vs += __shfl_xor(vs, m, 32);
    const float rstd = rsqrtf(vs * (1.0f / 64.0f) + LN_EPS);
    out[n * HID + c0] = d0 * rstd * ln_w[c0] + ln_b[c0];
    out[n * HID + c1] = d1 * rstd * ln_w[c1] + ln_b[c1];
}

extern "C" void kernel_launch(void* const* d_in, const int* in_sizes, int n_in,
                              void* d_out, int out_size, void* d_ws, size_t ws_size,
                              hipStream_t stream) {
    const float* x    = (const float*)d_in[0];
    const int*   ei   = (const int*)  d_in[1];   // [2,E]
    const float* ea   = (const float*)d_in[2];
    const float* Wl   = (const float*)d_in[3];
    const float* Wr   = (const float*)d_in[4];
    const float* We   = (const float*)d_in[5];
    const float* att  = (const float*)d_in[6];
    const float* bias = (const float*)d_in[7];
    const float* lnw  = (const float*)d_in[8];
    const float* lnb  = (const float*)d_in[9];
    const int* src = ei;
    const int* dst = ei + N_EDGES;
    float* out = (float*)d_out;

    // workspace layout
    float* ws     = (float*)d_ws;
    float* xl     = ws;                               // N*256
    float* xr     = xl     + (size_t)N_NODES * FDIM;  // N*256
    float* logits = xr     + (size_t)N_NODES * FDIM;  // E*4 (reused in-place for ex)
    float* amax   = logits + (size_t)N_EDGES * HEADS; // N*4
    float* denom  = amax   + (size_t)N_NODES * HEADS; // N*4

    // 0: init
    gat_init_kernel<<<(N_NODES * HID + 255) / 256, 256, 0, stream>>>(out, amax, denom);

    // 1: node projections (WMMA): 3125 row tiles x (16 col tiles x 2 matrices)
    gat_proj_kernel<<<dim3(N_NODES / 16, 32), 32, 0, stream>>>(x, Wl, Wr, xl, xr);

    // 2: fused edge GEMM + logits + atomic max: 50000 edge tiles x 4 heads
    gat_edge_logits_kernel<<<dim3(N_EDGES / 16, HEADS), 32, 0, stream>>>(
        ea, We, xl, xr, att, src, dst, logits, amax);

    // 3: exp + denom
    gat_exp_kernel<<<(N_EDGES * HEADS + 255) / 256, 256, 0, stream>>>(dst, amax, logits, denom);

    // 4: scatter aggregate (wave per edge, 8 edges per block)
    gat_aggregate_kernel<<<(N_EDGES + 7) / 8, 256, 0, stream>>>(xl, logits, denom, src, dst, out);

    // 5: residual + LayerNorm (wave per node, 8 nodes per block)
    gat_finalize_kernel<<<(N_NODES + 7) / 8, 256, 0, stream>>>(x, bias, lnw, lnb, out);
}